// Net_53163105190632
// MI455X (gfx1250) — compile-verified
//
#include <hip/hip_runtime.h>
#include <hip/hip_bf16.h>

#define NN 100000      // nodes
#define NE 1600000     // edges
#define D  128         // feature dim (D_IN == D_H)
#define DOUT 32
#define NG 128         // graphs
#define BN_EPS 1e-5f

typedef __attribute__((ext_vector_type(2))) float v2f;
typedef __attribute__((ext_vector_type(8))) float v8f;

__device__ __forceinline__ void atomAddF(float* p, float v) {
    unsafeAtomicAdd(p, v);   // lowers to global_atomic_add_f32 (no return)
}

// ---------------------------------------------------------------------------
// Edge scatter: agg[dst] += x[src].  32 threads per edge, float4 per thread.
// agg must be pre-initialized with x (gives h = x + agg).
// ---------------------------------------------------------------------------
__global__ void scatter_edges(const float* __restrict__ x,
                              const int* __restrict__ src,
                              const int* __restrict__ dst,
                              float* __restrict__ agg) {
    unsigned tid = blockIdx.x * blockDim.x + threadIdx.x;   // NE*32 threads exactly
    unsigned e = tid >> 5;
    unsigned c = (tid & 31u) * 4u;
    int s = src[e];
    int d = dst[e];
    const float4 v = *(const float4*)(x + (size_t)s * D + c);
    float* a = agg + (size_t)d * D + c;
    atomAddF(a + 0, v.x);
    atomAddF(a + 1, v.y);
    atomAddF(a + 2, v.z);
    atomAddF(a + 3, v.w);
}

// ---------------------------------------------------------------------------
// WMMA GEMM: C[M x 128] = A[M x 128] @ W[128 x 128] + bias (+ optional ReLU)
// One wave per 16x16 tile; grid sized exactly (M % 16 == 0), so EXEC == all 1s.
// Uses V_WMMA_F32_16X16X4_F32 (fp32, K-step 4).
// ---------------------------------------------------------------------------
__global__ void gemm_wmma_bias(const float* __restrict__ A,
                               const float* __restrict__ W,
                               const float* __restrict__ bias,
                               float* __restrict__ C,
                               int relu_out) {
    const int lane = threadIdx.x & 31;
    const int wave = threadIdx.x >> 5;
    const int tile = blockIdx.x * 8 + wave;   // 8 waves per block
    const int mt = tile >> 3;                 // N has 8 tiles of 16
    const int nt = tile & 7;

    const int half = lane >> 4;               // 0: lanes 0-15, 1: lanes 16-31
    const int l    = lane & 15;

    // A-fragment row for this lane (both halves cover M=0..15 of the tile)
    const float* __restrict__ arow = A + (size_t)(mt * 16 + l) * D;
    const int n = nt * 16 + l;                // B/C column for this lane

    v8f acc = {};
    #pragma unroll
    for (int k = 0; k < D; k += 4) {
        // A 16x4 fp32 layout: VGPR0 = K+0 (half0) / K+2 (half1); VGPR1 = K+1 / K+3
        v2f a = *(const v2f*)(arow + k + half * 2);
        // B 4x16 fp32 layout mirrors A: lanes hold column n, VGPR pair holds K split
        v2f b;
        b.x = W[(size_t)(k + half * 2 + 0) * D + n];
        b.y = W[(size_t)(k + half * 2 + 1) * D + n];
        acc = __builtin_amdgcn_wmma_f32_16x16x4_f32(
            /*neg_a=*/false, a, /*neg_b=*/false, b,
            /*c_mod=*/(short)0, acc, /*reuse_a=*/false, /*reuse_b=*/false);
    }

    // C layout: VGPR r -> row (r + 8*half), col n
    const float bn = bias[n];
    const int row0 = mt * 16 + half * 8;
    #pragma unroll
    for (int r = 0; r < 8; ++r) {
        float v = acc[r] + bn;
        if (relu_out) v = fmaxf(v, 0.0f);
        C[(size_t)(row0 + r) * D + n] = v;
    }
}

// ---------------------------------------------------------------------------
// BatchNorm column statistics: each block reduces 512 rows, 1 column/thread.
// ---------------------------------------------------------------------------
__global__ void bn_stats(const float* __restrict__ h,
                         float* __restrict__ sums,
                         float* __restrict__ sumsq) {
    const int col = threadIdx.x;                 // 128 threads
    int r0 = blockIdx.x * 512;
    int r1 = r0 + 512; if (r1 > NN) r1 = NN;
    float s = 0.f, s2 = 0.f;
    for (int r = r0; r < r1; ++r) {
        float v = h[(size_t)r * D + col];
        s += v;
        s2 = fmaf(v, v, s2);
    }
    atomAddF(&sums[col], s);
    atomAddF(&sumsq[col], s2);
}

__global__ void bn_finalize(const float* __restrict__ sums,
                            const float* __restrict__ sumsq,
                            const float* __restrict__ gamma,
                            const float* __restrict__ beta,
                            float* __restrict__ scale,
                            float* __restrict__ shift) {
    const int c = threadIdx.x;                   // 128 threads, 1 block
    const float inv_n = 1.0f / (float)NN;
    float mean = sums[c] * inv_n;
    float var  = sumsq[c] * inv_n - mean * mean;
    float sc   = gamma[c] * rsqrtf(var + BN_EPS);
    scale[c] = sc;
    shift[c] = beta[c] - mean * sc;
}

// Fused (h - mean)/sigma * gamma + beta, then ReLU, in place. float4 per thread.
__global__ void bn_apply_relu(float* __restrict__ h,
                              const float* __restrict__ scale,
                              const float* __restrict__ shift) {
    unsigned i = (blockIdx.x * blockDim.x + threadIdx.x) * 4u;  // NN*D floats exactly
    unsigned col = i & (D - 1);
    float4 v = *(float4*)(h + i);
    v.x = fmaxf(fmaf(v.x, scale[col + 0], shift[col + 0]), 0.f);
    v.y = fmaxf(fmaf(v.y, scale[col + 1], shift[col + 1]), 0.f);
    v.z = fmaxf(fmaf(v.z, scale[col + 2], shift[col + 2]), 0.f);
    v.w = fmaxf(fmaf(v.w, scale[col + 3], shift[col + 3]), 0.f);
    *(float4*)(h + i) = v;
}

// ---------------------------------------------------------------------------
// global_add_pool: pooled[batch[n]] += x[n].  32 threads/node, float4/thread.
// ---------------------------------------------------------------------------
__global__ void pool_nodes(const float* __restrict__ x,
                           const int* __restrict__ batch,
                           float* __restrict__ pooled) {
    unsigned tid = blockIdx.x * blockDim.x + threadIdx.x;   // NN*32 threads exactly
    unsigned node = tid >> 5;
    unsigned c = (tid & 31u) * 4u;
    int g = batch[node];
    const float4 v = *(const float4*)(x + (size_t)node * D + c);
    float* p = pooled + (size_t)g * D + c;
    atomAddF(p + 0, v.x);
    atomAddF(p + 1, v.y);
    atomAddF(p + 2, v.z);
    atomAddF(p + 3, v.w);
}

// ---------------------------------------------------------------------------
// Final MLP: one block per graph (128 threads). 4 MFLOP total -> latency bound.
// ---------------------------------------------------------------------------
__global__ void final_mlp(const float* __restrict__ pooled,
                          const float* __restrict__ W1,
                          const float* __restrict__ b1,
                          const float* __restrict__ W2,
                          const float* __restrict__ b2,
                          float* __restrict__ out) {
    __shared__ float p[D];
    __shared__ float hh[D];
    const int g = blockIdx.x;
    const int t = threadIdx.x;
    p[t] = pooled[(size_t)g * D + t];
    __syncthreads();
    float acc = b1[t];
    for (int k = 0; k < D; ++k) acc = fmaf(p[k], W1[(size_t)k * D + t], acc);
    hh[t] = fmaxf(acc, 0.f);
    __syncthreads();
    if (t < DOUT) {
        float o = b2[t];
        for (int k = 0; k < D; ++k) o = fmaf(hh[k], W2[(size_t)k * DOUT + t], o);
        out[(size_t)g * DOUT + t] = o;
    }
}

// ---------------------------------------------------------------------------
extern "C" void kernel_launch(void* const* d_in, const int* in_sizes, int n_in,
                              void* d_out, int out_size, void* d_ws, size_t ws_size,
                              hipStream_t stream) {
    const float* x0        = (const float*)d_in[0];
    const int*   edge      = (const int*)d_in[1];     // (2, NE): [0..NE)=src, [NE..2NE)=dst
    const int*   batch     = (const int*)d_in[2];
    const float* conv_W1   = (const float*)d_in[3];   // (4,128,128)
    const float* conv_b1   = (const float*)d_in[4];   // (4,128)
    const float* conv_g    = (const float*)d_in[5];
    const float* conv_be   = (const float*)d_in[6];
    const float* conv_W2   = (const float*)d_in[7];
    const float* conv_b2   = (const float*)d_in[8];
    const float* mlp_W1    = (const float*)d_in[9];
    const float* mlp_b1    = (const float*)d_in[10];
    const float* mlp_W2    = (const float*)d_in[11];
    const float* mlp_b2    = (const float*)d_in[12];

    const int* e_src = edge;
    const int* e_dst = edge + NE;

    const size_t NF = (size_t)NN * D;                 // 12.8M floats
    float* bufA  = (float*)d_ws;                      // aggregation buffer
    float* bufB  = bufA + NF;                         // hidden (pre/post BN)
    float* bufX  = bufB + NF;                         // layer output x
    float* sums  = bufX + NF;                         // [128]
    float* sumsq = sums + D;                          // [128]
    float* scale = sumsq + D;                         // [128]
    float* shift = scale + D;                         // [128]
    float* pooled = shift + D;                        // [128 x 128]

    const int gemm_blocks = (NN / 16) * (D / 16) / 8; // 6250 blocks of 8 waves
    const int scat_blocks = (NE * 32) / 256;          // 200000
    const int bnap_blocks = (int)(NF / 4 / 256);      // 12500
    const int stat_blocks = (NN + 511) / 512;         // 196
    const int pool_blocks = (NN * 32) / 256;          // 12500

    const float* xin = x0;
    for (int l = 0; l < 4; ++l) {
        // agg = x  (so scatter produces h = x + sum_{src->dst} x[src])
        hipMemcpyAsync(bufA, xin, NF * sizeof(float), hipMemcpyDeviceToDevice, stream);
        scatter_edges<<<scat_blocks, 256, 0, stream>>>(xin, e_src, e_dst, bufA);

        // h = agg @ W1 + b1
        gemm_wmma_bias<<<gemm_blocks, 256, 0, stream>>>(
            bufA, conv_W1 + (size_t)l * D * D, conv_b1 + (size_t)l * D, bufB, 0);

        // BatchNorm stats + ReLU
        hipMemsetAsync(sums, 0, 2 * D * sizeof(float), stream);  // sums + sumsq
        bn_stats<<<stat_blocks, D, 0, stream>>>(bufB, sums, sumsq);
        bn_finalize<<<1, D, 0, stream>>>(sums, sumsq,
                                         conv_g + (size_t)l * D, conv_be + (size_t)l * D,
                                         scale, shift);
        bn_apply_relu<<<bnap_blocks, 256, 0, stream>>>(bufB, scale, shift);

        // x = relu(h @ W2 + b2)
        gemm_wmma_bias<<<gemm_blocks, 256, 0, stream>>>(
            bufB, conv_W2 + (size_t)l * D * D, conv_b2 + (size_t)l * D, bufX, 1);

        xin = bufX;
    }

    // global_add_pool
    hipMemsetAsync(pooled, 0, (size_t)NG * D * sizeof(float), stream);
    pool_nodes<<<pool_blocks, 256, 0, stream>>>(bufX, batch, pooled);

    // final MLP -> logits (128 x 32)
    final_mlp<<<NG, D, 0, stream>>>(pooled, mlp_W1, mlp_b1, mlp_W2, mlp_b2,
                                    (float*)d_out);
}